// LIF_16484084483623
// MI455X (gfx1250) — compile-verified
//
#include <hip/hip_runtime.h>
#include <cstdint>

// LIF recurrence, T=5, state [B=32,C=32,H=64,W=64] (per reference; shapes fixed).
// Streaming, memory-bound: ~252 MB traffic -> ~10.8 us floor at 23.3 TB/s.
// CDNA5 path: GLOBAL_LOAD_ASYNC_TO_LDS_B128 (GVS mode, NT hint) pipeline with
// staged s_wait_asynccnt; recurrence state in VGPRs; weight sigmoids folded to
// native v_tanh_f32; clamps as v_med3_f32; NT B128 spike stores.

#define TSTEPS 5
#define BLOCK  256

// Native clang vector: accepted by __builtin_nontemporal_store, lowers to B128.
typedef __attribute__((ext_vector_type(4))) float v4f;

__device__ __forceinline__ float sigmoidf_fast(float x) {
    return 1.0f / (1.0f + __expf(-x));
}

__device__ __forceinline__ float fast_tanh(float x) {
#if __has_builtin(__builtin_amdgcn_tanhf)
    return __builtin_amdgcn_tanhf(x);   // v_tanh_f32 (gfx1250 trans op)
#else
    return tanhf(x);
#endif
}

__device__ __forceinline__ float clampf(float x, float lo, float hi) {
#if __has_builtin(__builtin_amdgcn_fmed3f)
    return __builtin_amdgcn_fmed3f(x, lo, hi);  // single v_med3_f32
#else
    return fminf(fmaxf(x, lo), hi);
#endif
}

template <int K>
__device__ __forceinline__ void wait_asynccnt() {
#if __has_builtin(__builtin_amdgcn_s_wait_asynccnt)
    __builtin_amdgcn_s_wait_asynccnt(K);
#else
    asm volatile("s_wait_asynccnt %0" ::"n"(K) : "memory");
#endif
}

// Async copy 16B global->LDS, GVS addressing: SGPR-pair base + 32-bit VGPR byte
// offset. NT hint: inputs are single-use, keep them out of temporal residency.
__device__ __forceinline__ void async_load_b128(uint32_t lds_addr, uint32_t byte_off,
                                                const float* sbase) {
    asm volatile("global_load_async_to_lds_b128 %0, %1, %2 th:TH_LOAD_NT"
                 :
                 : "v"(lds_addr), "v"(byte_off), "s"(sbase)
                 : "memory");
}

__global__ void __launch_bounds__(BLOCK) lif_async_kernel(
    const float* __restrict__ x_exc,
    const float* __restrict__ x_inh,
    const float* __restrict__ tau_param,
    float* __restrict__ out,
    int n4,            // number of float4 sites per timestep (N/4)
    int stride_elems)  // N = B*C*H*W, element stride between timesteps
{
    // 2 inputs * 5 timesteps * 256 threads * 16B = 40 KB LDS staging.
    __shared__ v4f stage[2 * TSTEPS * BLOCK];

    const int tid = threadIdx.x;
    const int gid = blockIdx.x * BLOCK + tid;
    if (gid >= n4) return;  // grid divides exactly; uniform per block anyway

    const long long e  = (long long)gid * 4;       // element offset in one timestep
    const uint32_t eb  = (uint32_t)gid * 16u;      // byte offset in one timestep
    const uint32_t tsb = (uint32_t)stride_elems * 4u;  // bytes per timestep slab

    // ---- Kick off all 10 async B128 loads (deep prefetch, ASYNCcnt = 10) ----
#pragma unroll
    for (int t = 0; t < TSTEPS; ++t) {
        const uint32_t off = eb + (uint32_t)t * tsb;   // < 2^31, GVS-safe
        const uint32_t le  = (uint32_t)(uintptr_t)&stage[(2 * t + 0) * BLOCK + tid];
        const uint32_t li  = (uint32_t)(uintptr_t)&stage[(2 * t + 1) * BLOCK + tid];
        async_load_b128(le, off, x_exc);
        async_load_b128(li, off, x_inh);
    }

    // ---- dt_tau: per-channel, C=32, H*W=4096 -> c = (e >> 12) & 31 ----
    const int c = ((int)(e >> 12)) & 31;
    const float dt = clampf(sigmoidf_fast(tau_param[c]), 0.05f, 1.0f);

    v4f mem  = {0.f, 0.f, 0.f, 0.f};
    v4f ema  = {0.f, 0.f, 0.f, 0.f};
    v4f gmod = {0.f, 0.f, 0.f, 0.f};

#pragma unroll
    for (int t = 0; t < TSTEPS; ++t) {
        // Async loads complete in issue order: after timestep t's pair lands,
        // ASYNCcnt <= 10 - 2*(t+1). Branches fold under full unroll.
        if      (t == 0) wait_asynccnt<8>();
        else if (t == 1) wait_asynccnt<6>();
        else if (t == 2) wait_asynccnt<4>();
        else if (t == 3) wait_asynccnt<2>();
        else             wait_asynccnt<0>();

        const v4f ge = stage[(2 * t + 0) * BLOCK + tid];
        const v4f gi = stage[(2 * t + 1) * BLOCK + tid];
        v4f s;

        // weight = 2*(sigmoid(0.5-ema) - sigmoid(ema-0.7))
        //        = tanh((0.5-ema)/2) - tanh((ema-0.7)/2)   (exact identity)
#define LIF_COMP(k)                                                              \
        {                                                                        \
            float m  = mem.k;                                                    \
            float em = ema.k;                                                    \
            float gm = gmod.k;                                                   \
            const float leak = clampf(0.7f - gm, 0.01f, 1.0f);                   \
            const float d = (ge.k * (1.5f - m) + gi.k * (-0.5f - m)              \
                             - leak * m) * dt;                                   \
            m += d;                                                              \
            const float sp = (m - 0.5f >= 0.0f) ? 1.0f : 0.0f;                   \
            em = 0.5f * em + 0.5f * sp;                                          \
            const float w = fast_tanh((0.5f - em) * 0.5f)                        \
                          - fast_tanh((em - 0.7f) * 0.5f);                       \
            gm = 0.9f * gm + 0.1f * w;                                           \
            m -= 0.5f * sp;                                                      \
            mem.k = m; ema.k = em; gmod.k = gm; s.k = sp;                        \
        }
        LIF_COMP(x) LIF_COMP(y) LIF_COMP(z) LIF_COMP(w)
#undef LIF_COMP

        // Spikes are write-once, never re-read: non-temporal B128 store.
        v4f* op = (v4f*)(out + (long long)t * stride_elems + e);
        __builtin_nontemporal_store(s, op);
    }
}

extern "C" void kernel_launch(void* const* d_in, const int* in_sizes, int n_in,
                              void* d_out, int out_size, void* d_ws, size_t ws_size,
                              hipStream_t stream) {
    (void)n_in; (void)d_ws; (void)ws_size; (void)out_size;

    const float* x_exc = (const float*)d_in[0];  // [T,B,C,H,W] f32
    const float* x_inh = (const float*)d_in[1];  // [T,B,C,H,W] f32
    const float* tau   = (const float*)d_in[2];  // [C,1,1]     f32
    float* out = (float*)d_out;                  // [T,B,C,H,W] f32

    const int T = TSTEPS;
    const int total = in_sizes[0];       // T*B*C*H*W = 20,971,520
    const int N = total / T;             // 4,194,304 elements per timestep
    const int n4 = N / 4;                // 1,048,576 float4 sites

    dim3 block(BLOCK);
    dim3 grid((n4 + BLOCK - 1) / BLOCK); // 4096 blocks, exact fit

    lif_async_kernel<<<grid, block, 0, stream>>>(x_exc, x_inh, tau, out, n4, N);
}